// SupCRLoss_84198538870869
// MI455X (gfx1250) — compile-verified
//
#include <hip/hip_runtime.h>
#include <math.h>

#define NN 2048
#define BB 1024
#define DD 512
#define TEMP 0.07f

typedef __attribute__((ext_vector_type(2))) float v2f;
typedef __attribute__((ext_vector_type(8))) float v8f;

// cf row i = features[b, v, :] with b = i % 1024, v = i / 1024 (features is [1024][2][512] row-major)
__device__ __forceinline__ const float* cf_row(const float* f, int i) {
    return f + ((size_t)((i & (BB - 1)) * 2 + (i >> 10))) * DD;
}

// ---------------- squared norms: sq[i] = dot(cf_i, cf_i) ----------------
__global__ void __launch_bounds__(64) sq_kernel(const float* __restrict__ feat,
                                                float* __restrict__ sq) {
    __shared__ float red[64];
    const int i = blockIdx.x;
    const float* row = cf_row(feat, i);
    float acc = 0.f;
    for (int k = threadIdx.x; k < DD; k += 64) {
        float x = row[k];
        acc = fmaf(x, x, acc);
    }
    red[threadIdx.x] = acc;
    __syncthreads();
    for (int s = 32; s > 0; s >>= 1) {
        if (threadIdx.x < s) red[threadIdx.x] += red[threadIdx.x + s];
        __syncthreads();
    }
    if (threadIdx.x == 0) sq[i] = red[0];
}

// ---------------- G = cf * cf^T via V_WMMA_F32_16X16X4_F32 ----------------
// One wave per 16x16 tile; 8 waves per block; 128x128 tiles total.
__global__ void __launch_bounds__(256) gram_wmma_kernel(const float* __restrict__ feat,
                                                        float* __restrict__ G) {
    const int wave = threadIdx.x >> 5;
    const int lane = threadIdx.x & 31;
    const int tile = blockIdx.x * 8 + wave;
    const int tI = tile >> 7;
    const int tJ = tile & 127;

    const int m    = lane & 15;   // row within tile (A: M index, B: N index)
    const int half = lane >> 4;   // half-wave selects K pair {0,1} vs {2,3}

    // ISA 32-bit A 16x4 layout: VGPR0 = K0 (lanes 0-15) / K2 (lanes 16-31), VGPR1 = K1 / K3.
    const float* pa = cf_row(feat, tI * 16 + m) + 2 * half;
    const float* pb = cf_row(feat, tJ * 16 + m) + 2 * half;

    v8f acc = {};
    for (int k = 0; k < DD; k += 4) {
        v2f a = *(const v2f*)(pa + k);   // 8B-aligned global_load_b64
        v2f b = *(const v2f*)(pb + k);
        acc = __builtin_amdgcn_wmma_f32_16x16x4_f32(
            /*neg_a=*/false, a, /*neg_b=*/false, b,
            /*c_mod=*/(short)0, acc, /*reuse_a=*/false, /*reuse_b=*/false);
    }

    // C/D layout: VGPR r -> M = r (lanes 0-15) or M = 8+r (lanes 16-31), N = lane%16
    const int row0 = tI * 16 + half * 8;
    const int col  = tJ * 16 + m;
#pragma unroll
    for (int r = 0; r < 8; ++r) G[(size_t)(row0 + r) * NN + col] = acc[r];
}

// ---------------- per-row: sort + suffix-scan + searchsorted in LDS ----------------
__global__ void __launch_bounds__(256) rowloss_kernel(const float* __restrict__ labels,
                                                      const float* __restrict__ G,
                                                      const float* __restrict__ sq,
                                                      float* __restrict__ rowOut) {
    __shared__ float s_d[NN];    // dis keys
    __shared__ float s_w[NN];    // w payload (becomes scan ping buffer)
    __shared__ int   s_ix[NN];   // original index payload (diag exclusion)
    __shared__ float s_suf[NN];  // scan pong buffer
    __shared__ float s_red[256];

    const int tid  = threadIdx.x;
    const int i    = blockIdx.x;
    const float yi = labels[i & (BB - 1)];
    const float si = sq[i];

    // Phase 1: build w / dis rows; accumulate sum of logits (diag logit = 0).
    float logit_sum = 0.f;
    for (int l = tid; l < NN; l += 256) {
        float g  = G[(size_t)i * NN + l];
        float d2 = fmaxf(si + sq[l] - 2.f * g, 0.f);
        float dist  = (d2 > 0.f) ? sqrtf(d2) : 0.f;
        float logit = -dist * (1.0f / TEMP);
        float wv;
        if (l == i) { logit = 0.f; wv = 0.f; }      // eye masks in reference
        else        { wv = expf(logit); }
        logit_sum += logit;
        s_d[l]  = fabsf(yi - labels[l & (BB - 1)]);
        s_w[l]  = wv;
        s_ix[l] = l;
    }
    __syncthreads();

    // Phase 2: bitonic sort ascending by dis, payloads follow.
    for (int k = 2; k <= NN; k <<= 1) {
        for (int j = k >> 1; j > 0; j >>= 1) {
            for (int e = tid; e < NN; e += 256) {
                int p = e ^ j;
                if (p > e) {
                    float de = s_d[e], dp = s_d[p];
                    bool up = ((e & k) == 0);
                    if (up ? (de > dp) : (de < dp)) {
                        s_d[e] = dp; s_d[p] = de;
                        float we = s_w[e]; s_w[e] = s_w[p]; s_w[p] = we;
                        int ie = s_ix[e]; s_ix[e] = s_ix[p]; s_ix[p] = ie;
                    }
                }
            }
            __syncthreads();
        }
    }

    // Phase 3: suffix sums of sorted w (Hillis-Steele, ping-pong).
    float* A  = s_w;
    float* Bp = s_suf;
    for (int off = 1; off < NN; off <<= 1) {
        for (int e = tid; e < NN; e += 256) {
            float v = A[e];
            if (e + off < NN) v += A[e + off];
            Bp[e] = v;
        }
        __syncthreads();
        float* t = A; A = Bp; Bp = t;
    }
    // A now holds suffix[p] = sum_{q>=p} w_sorted[q]

    // Phase 4: negative[j] = suffix[lower_bound(dis_j)]; accumulate log, skip j == i.
    float logneg_sum = 0.f;
    for (int e = tid; e < NN; e += 256) {
        if (s_ix[e] == i) continue;                 // diag: negative forced to 1 -> log 0
        float key = s_d[e];
        int lo = 0, hi = NN;                        // searchsorted side='left'
        while (lo < hi) {
            int mid = (lo + hi) >> 1;
            if (s_d[mid] < key) lo = mid + 1; else hi = mid;
        }
        logneg_sum += logf(A[lo]);
    }

    // Phase 5: block reductions, write per-row mean_log_prob_pos.
    __syncthreads();
    s_red[tid] = logit_sum;
    __syncthreads();
    for (int s = 128; s > 0; s >>= 1) {
        if (tid < s) s_red[tid] += s_red[tid + s];
        __syncthreads();
    }
    float tot_logit = s_red[0];
    __syncthreads();
    s_red[tid] = logneg_sum;
    __syncthreads();
    for (int s = 128; s > 0; s >>= 1) {
        if (tid < s) s_red[tid] += s_red[tid + s];
        __syncthreads();
    }
    if (tid == 0) rowOut[i] = (tot_logit - s_red[0]) / (float)(NN - 1);
}

// ---------------- final reduction: loss = -(T/BT) * mean(rowOut) ----------------
__global__ void __launch_bounds__(256) finalize_kernel(const float* __restrict__ rowOut,
                                                       float* __restrict__ out) {
    __shared__ float red[256];
    float acc = 0.f;
    for (int e = threadIdx.x; e < NN; e += 256) acc += rowOut[e];
    red[threadIdx.x] = acc;
    __syncthreads();
    for (int s = 128; s > 0; s >>= 1) {
        if (threadIdx.x < s) red[threadIdx.x] += red[threadIdx.x + s];
        __syncthreads();
    }
    if (threadIdx.x == 0) out[0] = -(TEMP / 0.07f) * red[0] / (float)NN;
}

extern "C" void kernel_launch(void* const* d_in, const int* in_sizes, int n_in,
                              void* d_out, int out_size, void* d_ws, size_t ws_size,
                              hipStream_t stream) {
    const float* feat   = (const float*)d_in[0];  // [1024, 2, 512] f32
    const float* labels = (const float*)d_in[1];  // [1024] f32

    float* G      = (float*)d_ws;                 // 2048*2048*4 = 16 MB
    float* sq     = G + (size_t)NN * NN;          // 8 KB
    float* rowOut = sq + NN;                      // 8 KB

    sq_kernel<<<NN, 64, 0, stream>>>(feat, sq);
    gram_wmma_kernel<<<NN, 256, 0, stream>>>(feat, G);
    rowloss_kernel<<<NN, 256, 0, stream>>>(labels, G, sq, rowOut);
    finalize_kernel<<<1, 256, 0, stream>>>(rowOut, (float*)d_out);
}